// NetworkInterpolationMLP_15092515078515
// MI455X (gfx1250) — compile-verified
//
#include <hip/hip_runtime.h>
#include <hip/hip_bf16.h>

typedef __attribute__((ext_vector_type(2))) float v2f;
typedef __attribute__((ext_vector_type(8))) float v8f;
typedef unsigned int u32;
typedef __attribute__((ext_vector_type(4))) unsigned int u32x4;
typedef __attribute__((ext_vector_type(8))) int i32x8;
typedef __attribute__((ext_vector_type(4))) int i32x4;

#define THETA      10.0f
#define T_SAMPLE   100
#define K_REF      32
#define NF1        410    // fc1 outputs
#define NF0        3072   // fc1 inputs
#define NF2        10     // fc2 outputs
#define NBATCH     64
#define KCH        64     // K-chunk staged per TDM transfer
#define NCH        (NF0 / KCH)          // 48 chunks
#define LDS_STRIDE 66     // 64 data dwords + 2 pad dwords (TDM pad feature)
#define S1_STRIDE  416    // 410 + 6 zero pad -> guard-free GEMM2 K loop

#if defined(__has_builtin)
#  if __has_builtin(__builtin_amdgcn_tensor_load_to_lds)
#    define HAVE_TDM 1
#  else
#    define HAVE_TDM 0
#  endif
#else
#  define HAVE_TDM 0
#endif

#if HAVE_TDM
// ---------------------------------------------------------------------------
// Tensor Data Mover: DMA a 2D fp32 tile (tile_x x tile_y, row stride
// `row_stride` elements) from global into LDS at byte offset `lds_addr`.
// D# per CDNA5 ISA:
//   group0: count=1 | lds_addr | global_addr[56:0] | type=2
//   group1: data_size=2 (4B), pad_enable=1, pad_interval=5 (every 64 dwords),
//           pad_amount=1 (2 dwords) -> LDS row stride 66 dwords (bank-friendly,
//           keeps 8B alignment for ds_load_b64 fragment reads).
//   tensor_dim* measured from the tile origin so rows/cols beyond the tensor
//   are zero-filled by the TDM OOB logic (handles the ragged 410-row B tile).
// ---------------------------------------------------------------------------
__device__ __forceinline__ void tdm_load_2d(u32 lds_addr, const float* gptr,
                                            u32 remain_x, u32 remain_y,
                                            u32 tile_x, u32 tile_y,
                                            u32 row_stride)
{
    unsigned long long ga = (unsigned long long)(size_t)gptr;
    u32x4 g0;
    g0.x = 1u;                                            // count=1, user mode
    g0.y = lds_addr;                                      // LDS byte address
    g0.z = (u32)ga;                                       // global_addr[31:0]
    g0.w = (u32)((ga >> 32) & 0x01FFFFFFu) | (2u << 30);  // addr[56:32] | type=2
    i32x8 g1;
    g1[0] = (int)((2u << 16) | (1u << 20) | (5u << 22) | (1u << 25));
    g1[1] = (int)((remain_x & 0xFFFFu) << 16);            // tensor_dim0[15:0]
    g1[2] = (int)((remain_x >> 16) | ((remain_y & 0xFFFFu) << 16));
    g1[3] = (int)((remain_y >> 16) | (tile_x << 16));     // tile_dim0
    g1[4] = (int)tile_y;                                  // tile_dim1 (tile_dim2=0)
    g1[5] = (int)row_stride;                              // tensor_dim0_stride lo
    g1[6] = 0;
    g1[7] = 0;
    i32x4 z4 = {0, 0, 0, 0};
#if __clang_major__ >= 23
    i32x8 z8 = {0, 0, 0, 0, 0, 0, 0, 0};
    __builtin_amdgcn_tensor_load_to_lds(g0, g1, z4, z4, z8, 0);
#else
    __builtin_amdgcn_tensor_load_to_lds(g0, g1, z4, z4, 0);
#endif
}
#endif // HAVE_TDM

// ---------------------------------------------------------------------------
// GEMM1: a1[64x410] = x[64x3072] * w1^T (w1 is [410x3072] row-major).
// One wave32 per 16x16 C tile.  A/B tiles staged in LDS by the TDM (double
// buffered, overlapped with WMMA via s_wait_tensorcnt), fragments read with
// ds_load_b64, accumulated with fp32 WMMA (precision matters: hard spike
// threshold downstream).
// ---------------------------------------------------------------------------
__global__ __launch_bounds__(32) void gemm1_wmma(const float* __restrict__ x,
                                                 const float* __restrict__ w,
                                                 float* __restrict__ out)
{
    __shared__ float lA[2][16 * LDS_STRIDE];
    __shared__ float lB[2][16 * LDS_STRIDE];

    const int tn   = blockIdx.x;          // 0..25
    const int tm   = blockIdx.y;          // 0..3
    const int lane = threadIdx.x;
    const int half = lane >> 4;
    const int l16  = lane & 15;
    const int mbase = tm * 16;
    const int nbase = tn * 16;
    const int bcol  = nbase + l16;
    const bool bvalid = (bcol < NF1);

#if HAVE_TDM
    tdm_load_2d((u32)(size_t)&lA[0][0], x + (size_t)mbase * NF0,
                NF0, (u32)(64 - mbase), KCH, 16, NF0);
    tdm_load_2d((u32)(size_t)&lB[0][0], w + (size_t)nbase * NF0,
                NF0, (u32)(NF1 - nbase), KCH, 16, NF0);
#endif

    v8f c = {};
    for (int ch = 0; ch < NCH; ++ch) {
        const int db = ch & 1;
#if HAVE_TDM
        if (ch + 1 < NCH) {
            const int k0n = (ch + 1) * KCH;
            tdm_load_2d((u32)(size_t)&lA[db ^ 1][0], x + (size_t)mbase * NF0 + k0n,
                        (u32)(NF0 - k0n), (u32)(64 - mbase), KCH, 16, NF0);
            tdm_load_2d((u32)(size_t)&lB[db ^ 1][0], w + (size_t)nbase * NF0 + k0n,
                        (u32)(NF0 - k0n), (u32)(NF1 - nbase), KCH, 16, NF0);
            __builtin_amdgcn_s_wait_tensorcnt(2);  // current chunk landed
        } else {
            __builtin_amdgcn_s_wait_tensorcnt(0);
        }
#else
        // Fallback: cooperative coalesced copy (LDS ops are in-order per wave).
        const int k0 = ch * KCH;
        for (int idx = lane; idx < 16 * KCH; idx += 32) {
            const int r  = idx >> 6;
            const int cc = idx & (KCH - 1);
            lA[db][r * LDS_STRIDE + cc] = x[(size_t)(mbase + r) * NF0 + k0 + cc];
            lB[db][r * LDS_STRIDE + cc] =
                (nbase + r < NF1) ? w[(size_t)(nbase + r) * NF0 + k0 + cc] : 0.0f;
        }
#endif
        const float* fA = &lA[db][l16 * LDS_STRIDE + half * 2];
        const float* fB = &lB[db][l16 * LDS_STRIDE + half * 2];
#pragma unroll
        for (int kk = 0; kk < KCH; kk += 4) {
            v2f a = *(const v2f*)(fA + kk);
            v2f b = *(const v2f*)(fB + kk);
            c = __builtin_amdgcn_wmma_f32_16x16x4_f32(
                    false, a, false, b, (short)0, c, false, false);
        }
    }

    if (bvalid) {
        const int rbase = mbase + half * 8;
#pragma unroll
        for (int r = 0; r < 8; ++r)
            out[(size_t)(rbase + r) * NF1 + bcol] = c[r];
    }
}

// ---------------------------------------------------------------------------
// Spike layer 1: psp of a time-constant input == a1 * cumsum(srm)[t].
// Writes s1 in (n, t, f) layout with row stride 416, zeroing the 6 pad
// columns so GEMM2's K loop needs no guards.
// ---------------------------------------------------------------------------
__global__ void spike1(const float* __restrict__ a1, float* __restrict__ s1)
{
    const int idx = blockIdx.x * blockDim.x + threadIdx.x;
    if (idx >= NBATCH * NF1) return;
    const int n = idx / NF1;
    const int f = idx - n * NF1;
    const float ubase = a1[idx];

    float rk[K_REF];
    {
        const float einv = __expf(-1.0f);
        float q = 1.0f;
#pragma unroll
        for (int i = 0; i < K_REF; ++i) { rk[i] = -2.0f * THETA * (float)(i + 1) * q; q *= einv; }
    }
    float buf[K_REF];
#pragma unroll
    for (int i = 0; i < K_REF; ++i) buf[i] = 0.0f;

    float S = 0.0f;
    float p = __expf(1.0f);
    const float d = __expf(-0.1f);

    for (int t = 0; t < T_SAMPLE; ++t) {
        S += ((float)t * 0.1f) * p;
        p *= d;
        const float m = fmaf(ubase, S, buf[0]);
        const float s = (m >= THETA) ? 1.0f : 0.0f;
#pragma unroll
        for (int i = 0; i < K_REF - 1; ++i) buf[i] = buf[i + 1];
        buf[K_REF - 1] = 0.0f;
#pragma unroll
        for (int i = 0; i < K_REF; ++i) buf[i] = fmaf(s, rk[i], buf[i]);
        const size_t row = (size_t)n * T_SAMPLE + t;
        s1[row * S1_STRIDE + f] = s;
        if (f < S1_STRIDE - NF1) s1[row * S1_STRIDE + NF1 + f] = 0.0f;  // zero pad
    }
}

// ---------------------------------------------------------------------------
// Prep: zero-padded w2p[16][416] so GEMM2's B path is guard-free.
// ---------------------------------------------------------------------------
__global__ void pad_w2(const float* __restrict__ w2, float* __restrict__ w2p)
{
    const int i = blockIdx.x * blockDim.x + threadIdx.x;
    if (i >= 16 * S1_STRIDE) return;
    const int col = i / S1_STRIDE;
    const int k   = i - col * S1_STRIDE;
    w2p[i] = (col < NF2 && k < NF1) ? w2[(size_t)col * NF1 + k] : 0.0f;
}

// ---------------------------------------------------------------------------
// GEMM2: a2[6400x16] = s1[6400x416] * w2p^T.  Guard-free fp32 WMMA K loop.
// ---------------------------------------------------------------------------
__global__ __launch_bounds__(32) void gemm2_wmma(const float* __restrict__ s1,
                                                 const float* __restrict__ w2p,
                                                 float* __restrict__ out)
{
    const int tm   = blockIdx.x;          // 0..399
    const int lane = threadIdx.x;
    const int half = lane >> 4;
    const int l16  = lane & 15;

    const float* aptr = s1  + (size_t)(tm * 16 + l16) * S1_STRIDE + half * 2;
    const float* bptr = w2p + (size_t)l16 * S1_STRIDE + half * 2;

    v8f c = {};
    for (int k = 0; k < S1_STRIDE; k += 4) {
        __builtin_prefetch(aptr + k + 128, 0, 1);
        v2f a = *(const v2f*)(aptr + k);
        v2f b = *(const v2f*)(bptr + k);
        c = __builtin_amdgcn_wmma_f32_16x16x4_f32(
                false, a, false, b, (short)0, c, false, false);
    }
    const int rbase = tm * 16 + half * 8;
#pragma unroll
    for (int r = 0; r < 8; ++r)
        out[(size_t)(rbase + r) * 16 + l16] = c[r];
}

// ---------------------------------------------------------------------------
// psp(a2) (100-tap causal conv) + spike layer 2; series staged in LDS.
// ---------------------------------------------------------------------------
__global__ __launch_bounds__(64) void psp_spike2(const float* __restrict__ a2,
                                                 float* __restrict__ out)
{
    __shared__ float srm[T_SAMPLE];
    __shared__ float ts[64][T_SAMPLE];

    const int tid = threadIdx.x;
    const int idx = blockIdx.x * 64 + tid;
    const int n = idx / NF2;
    const int o = idx - n * NF2;

    for (int t = tid; t < T_SAMPLE; t += 64)
        srm[t] = ((float)t * 0.1f) * __expf(1.0f - (float)t * 0.1f);
    for (int t = 0; t < T_SAMPLE; ++t)
        ts[tid][t] = a2[((size_t)n * T_SAMPLE + t) * 16 + o];
    __syncthreads();

    float rk[K_REF];
    {
        const float einv = __expf(-1.0f);
        float q = 1.0f;
#pragma unroll
        for (int i = 0; i < K_REF; ++i) { rk[i] = -2.0f * THETA * (float)(i + 1) * q; q *= einv; }
    }
    float buf[K_REF];
#pragma unroll
    for (int i = 0; i < K_REF; ++i) buf[i] = 0.0f;

    for (int t = 0; t < T_SAMPLE; ++t) {
        float u = 0.0f;
        for (int k = 0; k <= t; ++k)
            u = fmaf(srm[k], ts[tid][t - k], u);
        const float m = u + buf[0];
        const float s = (m >= THETA) ? 1.0f : 0.0f;
#pragma unroll
        for (int i = 0; i < K_REF - 1; ++i) buf[i] = buf[i + 1];
        buf[K_REF - 1] = 0.0f;
#pragma unroll
        for (int i = 0; i < K_REF; ++i) buf[i] = fmaf(s, rk[i], buf[i]);
        out[(size_t)n * (NF2 * T_SAMPLE) + (size_t)o * T_SAMPLE + t] = s;
    }
}

// ---------------------------------------------------------------------------
// Host-side launcher
// ---------------------------------------------------------------------------
extern "C" void kernel_launch(void* const* d_in, const int* in_sizes, int n_in,
                              void* d_out, int out_size, void* d_ws, size_t ws_size,
                              hipStream_t stream) {
    const float* inp = (const float*)d_in[0];   // (64,3,32,32) = 64x3072
    const float* w1  = (const float*)d_in[1];   // (410, 3072)
    const float* w2  = (const float*)d_in[2];   // (10, 410)
    float* out = (float*)d_out;                 // (64, 10, 100)

    char* ws = (char*)d_ws;
    const size_t A1_BYTES = (size_t)NBATCH * NF1 * sizeof(float);                    // 104,960
    const size_t S1_BYTES = (size_t)NBATCH * T_SAMPLE * S1_STRIDE * sizeof(float);   // 10,649,600
    const size_t A2_BYTES = (size_t)NBATCH * T_SAMPLE * 16 * sizeof(float);          // 409,600
    float* a1  = (float*)ws;
    float* s1  = (float*)(ws + A1_BYTES);
    float* a2  = (float*)(ws + A1_BYTES + S1_BYTES);
    float* w2p = (float*)(ws + A1_BYTES + S1_BYTES + A2_BYTES);

    pad_w2<<<(16 * S1_STRIDE + 255) / 256, 256, 0, stream>>>(w2, w2p);

    dim3 g1(26, 4);
    gemm1_wmma<<<g1, 32, 0, stream>>>(inp, w1, a1);

    const int nNeurons1 = NBATCH * NF1;
    spike1<<<(nNeurons1 + 255) / 256, 256, 0, stream>>>(a1, s1);

    gemm2_wmma<<<400, 32, 0, stream>>>(s1, w2p, a2);

    psp_spike2<<<10, 64, 0, stream>>>(a2, out);
}